// GlobalLocalCrossAttention_5506148073766
// MI455X (gfx1250) — compile-verified
//
#include <hip/hip_runtime.h>
#include <hip/hip_bf16.h>

typedef __attribute__((ext_vector_type(2))) float v2f;
typedef __attribute__((ext_vector_type(4))) float v4f;
typedef __attribute__((ext_vector_type(8))) float v8f;
typedef __attribute__((ext_vector_type(4))) unsigned int u32x4;
typedef __attribute__((ext_vector_type(8))) int i32x8;
typedef __attribute__((ext_vector_type(4))) int i32x4;

// Problem constants (from reference)
constexpr int Bn = 32;    // batch
constexpr int N1 = 197;   // tokens (incl CLS)
constexpr int Dm = 768;   // model dim
constexpr int Lc = 12;    // layers
constexpr int Hh = 12;    // heads
constexpr int HD = 64;    // head dim
constexpr int Rr = 19;    // top-k tokens = int(196*0.1)

#if __has_builtin(__builtin_amdgcn_tensor_load_to_lds)
#define HAVE_TDM 1
#else
#define HAVE_TDM 0
#endif

#if HAVE_TDM
// 2D TDM tile load: tile_h rows x 64 floats, global row stride in floats.
// D# pad fields insert 1 DWORD after every 64 DWORDs -> LDS row stride 65.
__device__ inline void tdm_load_tile64(void* lds_dst, const void* gsrc,
                                       int tile_h, int row_stride)
{
    const unsigned long long ga = (unsigned long long)(size_t)gsrc;
    const unsigned lds = (unsigned)(size_t)lds_dst;   // low 32 bits = LDS byte addr
    u32x4 g0;
    g0.x = 1u;                                        // count=1 (valid descriptor)
    g0.y = lds;                                       // lds_addr
    g0.z = (unsigned)(ga & 0xFFFFFFFFu);              // global_addr[31:0]
    g0.w = (unsigned)((ga >> 32) & 0x01FFFFFFu) | 0x80000000u; // addr[56:32]|type=2
    i32x8 g1;
    g1[0] = (2 << 16) | (1 << 20) | (5 << 22);        // 4B elems, pad_en, every 64 DW, +1 DW
    g1[1] = (64 & 0xFFFF) << 16;                      // tensor_dim0 = 64 (lo16)
    g1[2] = (tile_h & 0xFFFF) << 16;                  // dim0 hi=0 | tensor_dim1 lo16
    g1[3] = (64 & 0xFFFF) << 16;                      // dim1 hi=0 | tile_dim0 = 64
    g1[4] = (tile_h & 0xFFFF);                        // tile_dim1 | tile_dim2=0
    g1[5] = row_stride;                               // tensor_dim0_stride lo32
    g1[6] = 0;
    g1[7] = 0;
    const i32x4 z4 = {0, 0, 0, 0};                    // groups 2/3 unused (2D)
    const i32x8 z8 = {0, 0, 0, 0, 0, 0, 0, 0};
    __builtin_amdgcn_tensor_load_to_lds(g0, g1, z4, z4, z8, 0);
}
#endif

// ---------------------------------------------------------------------------
// Pass 1: head-mean of attn_stack + row sums (rowsum = 1 + sum_j meanA[i,j]).
// One block per (l,b,i) row; coalesced single pass over the 715 MB tensor.
// ---------------------------------------------------------------------------
__global__ __launch_bounds__(256) void mean_rowsum_kernel(
    const float* __restrict__ attn, float* __restrict__ meanA,
    float* __restrict__ rowsum)
{
    const int i = blockIdx.x, b = blockIdx.y, l = blockIdx.z;
    const int tid = threadIdx.x;
    __shared__ float red[256];
    float v = 0.f;
    if (tid < N1) {
        const size_t base = (((size_t)l * Bn + b) * Hh) * (size_t)(N1 * N1)
                            + (size_t)i * N1 + tid;
        float s = 0.f;
        #pragma unroll
        for (int h = 0; h < Hh; ++h) s += attn[base + (size_t)h * N1 * N1];
        v = s * (1.f / (float)Hh);
        meanA[(((size_t)l * Bn + b) * N1 + i) * N1 + tid] = v;
    }
    red[tid] = v;
    __syncthreads();
    for (int s = 128; s > 0; s >>= 1) {
        if (tid < s) red[tid] += red[tid + s];
        __syncthreads();
    }
    if (tid == 0) rowsum[((size_t)l * Bn + b) * N1 + i] = 1.f + red[0];
}

// ---------------------------------------------------------------------------
// Fused rollout chain + top-k + gather. One block per batch; u/w in LDS.
// Uses row-0 propagation: row0(A[11]...A[0]) = row0(A[11]) @ A[10] @ ... @ A[0]
// with normalization folded in: u' = w + w @ meanA, w = u / rowsum.
// meanA (59.6 MB) is L2-resident (192 MB L2) after pass 1.
// ---------------------------------------------------------------------------
__global__ __launch_bounds__(256) void rollout_topk_gather_kernel(
    const float* __restrict__ meanA, const float* __restrict__ rowsum,
    const float* __restrict__ x, float* __restrict__ x_local)
{
    const int b = blockIdx.x, tid = threadIdx.x;
    __shared__ float u[256];
    __shared__ float w[256];
    __shared__ float rv[256];
    __shared__ int   ri[256];
    __shared__ int   sel[Rr];

    // init: normalized row 0 of layer L-1
    {
        const int l = Lc - 1;
        const float rs = rowsum[((size_t)l * Bn + b) * N1];
        if (tid < N1)
            u[tid] = (meanA[(((size_t)l * Bn + b) * N1) * N1 + tid]
                      + (tid == 0 ? 1.f : 0.f)) / rs;
    }
    for (int l = Lc - 2; l >= 0; --l) {
        __syncthreads();
        if (tid < N1) w[tid] = u[tid] / rowsum[((size_t)l * Bn + b) * N1 + tid];
        __syncthreads();
        if (tid < N1) {
            const float* mp = meanA + (((size_t)l * Bn + b) * N1) * N1 + tid;
            float acc = w[tid];
            int i = 0;
            for (; i + 4 <= N1; i += 4) {
                acc += w[i + 0] * mp[(size_t)(i + 0) * N1];
                acc += w[i + 1] * mp[(size_t)(i + 1) * N1];
                acc += w[i + 2] * mp[(size_t)(i + 2) * N1];
                acc += w[i + 3] * mp[(size_t)(i + 3) * N1];
            }
            for (; i < N1; ++i) acc += w[i] * mp[(size_t)i * N1];
            u[tid] = acc;   // safe: loop reads only w[]
        }
        __syncthreads();
    }
    // top-19 of u[1..196], descending, lower-index tie-break (lax.top_k order)
    if (tid < 196) w[tid] = u[tid + 1];   // reuse w as mutable value array
    __syncthreads();
    for (int r = 0; r < Rr; ++r) {
        float v = -3.4e38f; int idx = 0x7fffffff;
        if (tid < 196) { v = w[tid]; idx = tid; }
        rv[tid] = v; ri[tid] = idx;
        __syncthreads();
        for (int s = 128; s > 0; s >>= 1) {
            if (tid < s) {
                if (rv[tid + s] > rv[tid] ||
                    (rv[tid + s] == rv[tid] && ri[tid + s] < ri[tid])) {
                    rv[tid] = rv[tid + s]; ri[tid] = ri[tid + s];
                }
            }
            __syncthreads();
        }
        if (tid == 0) { sel[r] = ri[0]; w[ri[0]] = -3.4e38f; }
        __syncthreads();
    }
    // gather x_local[b,r,:] = x[b, sel[r]+1, :]
    for (int idx = tid; idx < Rr * Dm; idx += 256) {
        const int r = idx / Dm, c = idx % Dm;
        x_local[((size_t)b * Rr + r) * Dm + c] =
            x[((size_t)b * N1 + sel[r] + 1) * Dm + c];
    }
}

// ---------------------------------------------------------------------------
// fp32 WMMA GEMM (NT): C[m,n] = sum_k A[m,k]*Bw[n,k] (+ bias[n])
// 256 threads = 8 waves; 64x128 block tile; each wave: 16x64 strip with four
// v_wmma_f32_16x16x4_f32 accumulators (A fragment reused 4x).
// LDS: As stride 17 (odd -> conflict-free), Bs stride 144 (==16 mod 32 ->
// the two half-wave K rows hit disjoint bank sets).
// ---------------------------------------------------------------------------
__global__ __launch_bounds__(256) void gemm_nt_wmma(
    const float* __restrict__ A, const float* __restrict__ Bw,
    const float* __restrict__ bias, float* __restrict__ C,
    int M, int N, int K)
{
    constexpr int ASTR = 17, BSTR = 144;
    __shared__ float As[64 * ASTR];
    __shared__ float Bs[16 * BSTR];
    const int tid  = threadIdx.x;
    const int wave = tid >> 5;
    const int lane = tid & 31;
    const int wm   = wave & 3;     // 4 row tiles of 16
    const int wn   = wave >> 2;    // 2 col groups of 64
    const int m0   = blockIdx.y * 64;
    const int n0   = blockIdx.x * 128;
    const int half = lane >> 4;
    const int l16  = lane & 15;

    v8f acc0 = {}, acc1 = {}, acc2 = {}, acc3 = {};

    for (int k0 = 0; k0 < K; k0 += 16) {
        {   // A panel: 64 rows x 16 k
            const int r  = tid >> 2;
            const int kk = (tid & 3) * 4;
            const int gm = m0 + r;
            v4f val = {};
            if (gm < M) val = *(const v4f*)(A + (size_t)gm * K + k0 + kk);
            As[r * ASTR + kk + 0] = val.x; As[r * ASTR + kk + 1] = val.y;
            As[r * ASTR + kk + 2] = val.z; As[r * ASTR + kk + 3] = val.w;
        }
        {   // B panel transposed to [k][n]: Bs[kk][n] = Bw[(n0+n)*K + k0+kk]
            const int n   = tid >> 1;            // 0..127
            const int kk0 = (tid & 1) * 8;       // 0 or 8
            const int gn  = n0 + n;
            v4f v0 = {}, v1 = {};
            if (gn < N) {
                v0 = *(const v4f*)(Bw + (size_t)gn * K + k0 + kk0);
                v1 = *(const v4f*)(Bw + (size_t)gn * K + k0 + kk0 + 4);
            }
            Bs[(kk0 + 0) * BSTR + n] = v0.x; Bs[(kk0 + 1) * BSTR + n] = v0.y;
            Bs[(kk0 + 2) * BSTR + n] = v0.z; Bs[(kk0 + 3) * BSTR + n] = v0.w;
            Bs[(kk0 + 4) * BSTR + n] = v1.x; Bs[(kk0 + 5) * BSTR + n] = v1.y;
            Bs[(kk0 + 6) * BSTR + n] = v1.z; Bs[(kk0 + 7) * BSTR + n] = v1.w;
        }
        __syncthreads();
        #pragma unroll
        for (int ks = 0; ks < 16; ks += 4) {
            // A frag 16x4: lanes 0-15 -> K = ks+{0,1}; lanes 16-31 -> ks+{2,3}
            const int am = (wm * 16 + l16) * ASTR + ks + half * 2;
            v2f a; a.x = As[am]; a.y = As[am + 1];
            // B frag 4x16: VGPR0 holds rows ks/ks+1 across the two half-waves
            const int bb = wn * 64 + l16;
            v2f b0, b1, b2, b3;
            const int r0 = (ks + half) * BSTR, r1 = (ks + 2 + half) * BSTR;
            b0.x = Bs[r0 + bb];      b0.y = Bs[r1 + bb];
            b1.x = Bs[r0 + bb + 16]; b1.y = Bs[r1 + bb + 16];
            b2.x = Bs[r0 + bb + 32]; b2.y = Bs[r1 + bb + 32];
            b3.x = Bs[r0 + bb + 48]; b3.y = Bs[r1 + bb + 48];
            acc0 = __builtin_amdgcn_wmma_f32_16x16x4_f32(false, a, false, b0,
                                                         (short)0, acc0, false, false);
            acc1 = __builtin_amdgcn_wmma_f32_16x16x4_f32(false, a, false, b1,
                                                         (short)0, acc1, false, false);
            acc2 = __builtin_amdgcn_wmma_f32_16x16x4_f32(false, a, false, b2,
                                                         (short)0, acc2, false, false);
            acc3 = __builtin_amdgcn_wmma_f32_16x16x4_f32(false, a, false, b3,
                                                         (short)0, acc3, false, false);
        }
        __syncthreads();
    }
    // C/D layout: VGPR v: lanes 0-15 -> (M=v, N=lane); lanes 16-31 -> (M=v+8)
    #pragma unroll
    for (int v = 0; v < 8; ++v) {
        const int gm = m0 + wm * 16 + v + half * 8;
        if (gm < M) {
            const int gn = n0 + wn * 64 + l16;
            float* cp = C + (size_t)gm * N + gn;
            cp[0]  = acc0[v] + (bias ? bias[gn]      : 0.f);
            cp[16] = acc1[v] + (bias ? bias[gn + 16] : 0.f);
            cp[32] = acc2[v] + (bias ? bias[gn + 32] : 0.f);
            cp[48] = acc3[v] + (bias ? bias[gn + 48] : 0.f);
        }
    }
}

// ---------------------------------------------------------------------------
// 19x197 attention per (b,h): scores -> softmax -> @V fused in LDS.
// V slice (197x64, row stride 768) staged to LDS by the Tensor Data Mover
// (row-padded to stride 65 via D# pad fields -> conflict-free reuse x19).
// ---------------------------------------------------------------------------
__global__ __launch_bounds__(256) void attn_kernel(
    const float* __restrict__ Q, const float* __restrict__ Kp,
    const float* __restrict__ Vp, float* __restrict__ out)
{
    const int h = blockIdx.x, b = blockIdx.y, tid = threadIdx.x;
    __shared__ float qs[Rr * HD];
    __shared__ float ps[Rr][200];
    __shared__ float Vs[N1 * 65 + 4];
    __shared__ float rmax[Rr], rinv[Rr];

    const float* vsrc = Vp + ((size_t)b * N1) * Dm + h * HD;
#if HAVE_TDM
    if (tid < 32) {   // wave 0 issues the tensor DMA for the whole block
        tdm_load_tile64(Vs, vsrc, N1, Dm);
        __builtin_amdgcn_s_wait_tensorcnt(0);
    }
#else
    for (int idx = tid; idx < N1 * HD; idx += 256) {
        const int n = idx >> 6, d = idx & 63;
        Vs[n * 65 + d] = vsrc[(size_t)n * Dm + d];
    }
#endif
    for (int idx = tid; idx < Rr * HD; idx += 256) {
        const int r = idx / HD, d = idx % HD;
        qs[idx] = Q[((size_t)b * Rr + r) * Dm + h * HD + d];
    }
    __syncthreads();
    const float scale = 0.125f;  // 64^-0.5
    for (int idx = tid; idx < Rr * N1; idx += 256) {
        const int r = idx / N1, n = idx % N1;
        const float* kr = Kp + ((size_t)b * N1 + n) * Dm + h * HD;
        float s = 0.f;
        #pragma unroll 8
        for (int d = 0; d < HD; ++d) s += qs[r * HD + d] * kr[d];
        ps[r][n] = s * scale;
    }
    __syncthreads();
    if (tid < Rr) {
        float m = -3.4e38f;
        for (int n = 0; n < N1; ++n) m = fmaxf(m, ps[tid][n]);
        float ssum = 0.f;
        for (int n = 0; n < N1; ++n) ssum += __expf(ps[tid][n] - m);
        rmax[tid] = m; rinv[tid] = 1.f / ssum;
    }
    __syncthreads();
    for (int idx = tid; idx < Rr * N1; idx += 256) {
        const int r = idx / N1, n = idx % N1;
        ps[r][n] = __expf(ps[r][n] - rmax[r]) * rinv[r];
    }
    __syncthreads();
    for (int idx = tid; idx < Rr * HD; idx += 256) {
        const int r = idx / HD, d = idx % HD;
        float acc = 0.f;
        for (int n = 0; n < N1; ++n) acc += ps[r][n] * Vs[n * 65 + d];
        out[((size_t)b * Rr + r) * Dm + h * HD + d] = acc;
    }
}

extern "C" void kernel_launch(void* const* d_in, const int* in_sizes, int n_in,
                              void* d_out, int out_size, void* d_ws, size_t ws_size,
                              hipStream_t stream)
{
    const float* x       = (const float*)d_in[0];  // (32,197,768)
    const float* attn_st = (const float*)d_in[1];  // (12,32,12,197,197)
    const float* q_w     = (const float*)d_in[2];  // (768,768)
    const float* k_w     = (const float*)d_in[3];
    const float* v_w     = (const float*)d_in[4];
    const float* proj_w  = (const float*)d_in[5];
    const float* proj_b  = (const float*)d_in[6];
    float* out           = (float*)d_out;          // (32,19,768)

    // Workspace partition (static, deterministic)
    size_t off = 0;
    auto alloc = [&](size_t bytes) -> void* {
        void* p = (char*)d_ws + off;
        off += (bytes + 255) & ~(size_t)255;
        return p;
    };
    float* meanA   = (float*)alloc((size_t)Lc * Bn * N1 * N1 * 4);   // 59.6 MB
    float* rowsum  = (float*)alloc((size_t)Lc * Bn * N1 * 4);
    float* Kout    = (float*)alloc((size_t)Bn * N1 * Dm * 4);        // 19.4 MB
    float* Vout    = (float*)alloc((size_t)Bn * N1 * Dm * 4);        // 19.4 MB
    float* xloc    = (float*)alloc((size_t)Bn * Rr * Dm * 4);
    float* Qout    = (float*)alloc((size_t)Bn * Rr * Dm * 4);
    float* attnout = (float*)alloc((size_t)Bn * Rr * Dm * 4);

    // 1) head-mean + row sums (single pass over 715 MB; HBM-bound floor)
    mean_rowsum_kernel<<<dim3(N1, Bn, Lc), 256, 0, stream>>>(attn_st, meanA, rowsum);

    // 2) fused rollout row-0 chain + top-19 + gather (L2-resident meanA)
    rollout_topk_gather_kernel<<<Bn, 256, 0, stream>>>(meanA, rowsum, x, xloc);

    // 3) projections via fp32 WMMA (NT GEMM: Y = X @ W^T)
    const int MKV = Bn * N1;   // 6304
    const int MQ  = Bn * Rr;   // 608
    dim3 gKV(Dm / 128, (MKV + 63) / 64);
    dim3 gQ (Dm / 128, (MQ  + 63) / 64);
    gemm_nt_wmma<<<gKV, 256, 0, stream>>>(x,    k_w, nullptr, Kout, MKV, Dm, Dm);
    gemm_nt_wmma<<<gKV, 256, 0, stream>>>(x,    v_w, nullptr, Vout, MKV, Dm, Dm);
    gemm_nt_wmma<<<gQ,  256, 0, stream>>>(xloc, q_w, nullptr, Qout, MQ,  Dm, Dm);

    // 4) per-(b,h) attention (TDM-staged V), then output projection to d_out
    attn_kernel<<<dim3(Hh, Bn), 256, 0, stream>>>(Qout, Kout, Vout, attnout);
    gemm_nt_wmma<<<gQ,  256, 0, stream>>>(attnout, proj_w, proj_b, out, MQ, Dm, Dm);
}